// LpmpGRLayer_50053548867788
// MI455X (gfx1250) — compile-verified
//
#include <hip/hip_runtime.h>
#include <hip/hip_bf16.h>
#include <math.h>

// ---------------------------------------------------------------------------
// Geometry
// ---------------------------------------------------------------------------
#define IMG_H   512
#define IMG_W   512
#define CPT     8              // output columns per thread (8-col strip)
#define RPT     8              // output rows per thread
#define BANDS   4              // 256 threads = 64 col-threads x 4 row bands
#define TILE_ROWS (RPT * BANDS)        // 32 rows per block
#define TILES_PER_IMG (IMG_H / TILE_ROWS)  // 16

typedef __attribute__((ext_vector_type(2))) float v2f;
typedef __attribute__((ext_vector_type(8))) float v8f;

// ---------------------------------------------------------------------------
// Kernel 0: zero the 96x(sum,sumsq) accumulators in d_ws (re-done every call).
// ---------------------------------------------------------------------------
__global__ __launch_bounds__(256) void zero_ws_kernel(float* __restrict__ ws, int n) {
    int i = blockIdx.x * blockDim.x + threadIdx.x;
    if (i < n) ws[i] = 0.0f;
}

// ---------------------------------------------------------------------------
// Register-rolling helpers for kernel 1 (no LDS in the hot loop).
// Thread ct owns output cols [8*ct, 8*ct+8); its x window w[0..15] covers
// cols 8*ct-4 .. 8*ct+11 (4 aligned float4 chunks).
// ---------------------------------------------------------------------------
__device__ __forceinline__ void loadx(float w[16], const float* __restrict__ xi,
                                      int m, int ct) {
    if (m < 0 || m >= IMG_H) {
#pragma unroll
        for (int j = 0; j < 16; ++j) w[j] = 0.0f;
        return;
    }
    const float4* row = reinterpret_cast<const float4*>(xi + (size_t)m * IMG_W);
#pragma unroll
    for (int j = 0; j < 4; ++j) {
        int cidx = ct * 2 - 1 + j;                 // float4 chunk index
        float4 v = make_float4(0.f, 0.f, 0.f, 0.f);
        if (cidx >= 0 && cidx < IMG_W / 4) v = row[cidx];
        w[4 * j + 0] = v.x; w[4 * j + 1] = v.y;
        w[4 * j + 2] = v.z; w[4 * j + 3] = v.w;
    }
}

// Sobel magnitude of row m (separable), then horizontal 3-sum -> h[0..7].
// mag is forced to 0 outside the image (the avg-pool zero-pads MAG, not x).
__device__ __forceinline__ void magh(float h[CPT], int m, int ct,
                                     const float xm1[16], const float x0[16],
                                     const float xp1[16]) {
    if (m < 0 || m >= IMG_H) {
#pragma unroll
        for (int j = 0; j < CPT; ++j) h[j] = 0.0f;
        return;
    }
    // window cols used: w[2..13]  (mag cols 8*ct-1 .. 8*ct+8)
    float s[12], d[12];
#pragma unroll
    for (int c = 0; c < 12; ++c) {
        int wc = c + 2;
        s[c] = fmaf(2.0f, x0[wc], xm1[wc] + xp1[wc]);  // vertical [1,2,1]
        d[c] = xm1[wc] - xp1[wc];                      // vertical [1,0,-1]
    }
    float mg[CPT + 2];
#pragma unroll
    for (int k = 0; k < CPT + 2; ++k) {
        float gx = s[k] - s[k + 2];                        // horizontal [1,0,-1]
        float gy = fmaf(2.0f, d[k + 1], d[k] + d[k + 2]);  // horizontal [1,2,1]
        float v  = sqrtf(fmaf(gx, gx, fmaf(gy, gy, 1e-6f)));
        int a = 8 * ct - 1 + k;                            // absolute column
        if (a < 0 || a >= IMG_W) v = 0.0f;                 // pool zero-pad on mag
        mg[k] = v;
    }
#pragma unroll
    for (int j = 0; j < CPT; ++j) h[j] = mg[j] + mg[j + 1] + mg[j + 2];
}

__device__ __forceinline__ void copy16(float dst[16], const float src[16]) {
#pragma unroll
    for (int j = 0; j < 16; ++j) dst[j] = src[j];
}
__device__ __forceinline__ void copy8(float dst[CPT], const float src[CPT]) {
#pragma unroll
    for (int j = 0; j < CPT; ++j) dst[j] = src[j];
}

// ---------------------------------------------------------------------------
// Kernel 1: fused Sobel magnitude -> 3x3 avg pool (count_include_pad) ->
//           per-(b,c) sum / sumsq. Fully register-rolling: the only LDS is a
//           16-float reduction scratch, so the kernel rides the HBM roofline
//           (~100 MB @ 23.3 TB/s ~= 4.3 us) instead of being LDS-issue bound.
// Grid: 96 images * 16 row-tiles (32 rows); block: 256 threads (8 wave32).
// ---------------------------------------------------------------------------
__global__ __launch_bounds__(256) void sobel_pool_stats_kernel(
    const float* __restrict__ x, float* __restrict__ ws) {

    const int tid  = threadIdx.x;
    const int ct   = tid & 63;                  // column strip [8*ct, 8*ct+8)
    const int band = tid >> 6;                  // 0..3 row band within tile
    const int tile = blockIdx.x & (TILES_PER_IMG - 1);
    const int img  = blockIdx.x / TILES_PER_IMG;       // 0..95 = b*3+c
    const int r0   = tile * TILE_ROWS + band * RPT;    // first output row
    const float* xi = x + (size_t)img * IMG_H * IMG_W;

    float xA[16], xB[16], xC[16];
    float h0[CPT], h1[CPT], h2[CPT];

    // Prologue: mag h-sum rows r0-1 and r0.
    loadx(xA, xi, r0 - 2, ct);
    loadx(xB, xi, r0 - 1, ct);
    loadx(xC, xi, r0,     ct);
    magh(h0, r0 - 1, ct, xA, xB, xC);
    copy16(xA, xB); copy16(xB, xC);
    loadx(xC, xi, r0 + 1, ct);
    magh(h1, r0, ct, xA, xB, xC);

    float s = 0.0f, s2 = 0.0f;
#pragma unroll
    for (int i = 0; i < RPT; ++i) {
        copy16(xA, xB); copy16(xB, xC);
        loadx(xC, xi, r0 + i + 2, ct);
        magh(h2, r0 + i + 1, ct, xA, xB, xC);     // mag h-sums of row r+1
#pragma unroll
        for (int j = 0; j < CPT; ++j) {
            float p = (h0[j] + h1[j] + h2[j]) * (1.0f / 9.0f);
            s  += p;
            s2  = fmaf(p, p, s2);
        }
        copy8(h0, h1); copy8(h1, h2);
    }

    // ---- wave shuffle reduce, tiny LDS hop across the 8 waves --------------
#pragma unroll
    for (int off = 16; off > 0; off >>= 1) {
        s  += __shfl_down(s,  off, 32);
        s2 += __shfl_down(s2, off, 32);
    }
    __shared__ float rs[8], rs2[8];
    if ((tid & 31) == 0) { rs[tid >> 5] = s; rs2[tid >> 5] = s2; }
    __syncthreads();
    if (tid == 0) {
        float S = 0.0f, S2 = 0.0f;
#pragma unroll
        for (int w = 0; w < 8; ++w) { S += rs[w]; S2 += rs2[w]; }
        atomicAdd(&ws[img * 2 + 0], S);
        atomicAdd(&ws[img * 2 + 1], S2);
    }
}

// ---------------------------------------------------------------------------
// Kernel 2: mean/std -> feats[32,8] (K padded) -> WMMA f32 MLP.
// Single wave32 workgroup; EXEC all-ones as WMMA requires.
//   GEMM1: feats[32x8] x w1t[8x32]  (2x2 tiles, 2 k-steps)  + bias + relu
//   GEMM2: h[32x32]    x w2t[32x64] (2x4 tiles, 8 k-steps)  + bias
// V_WMMA_F32_16X16X4_F32 layouts (ISA 7.12.2, 32-bit):
//   A 16x4 : VGPR0 holds K=2*half, VGPR1 holds K=2*half+1 (half = lane>>4)
//   B 4x16 : mirrored (row striped across lanes within a VGPR)
//   C 16x16: VGPR v -> M = v + 8*half, N = lane&15
// ---------------------------------------------------------------------------
__global__ __launch_bounds__(32) void mlp_wmma_kernel(
    const float* __restrict__ ws,
    const float* __restrict__ w1, const float* __restrict__ b1,
    const float* __restrict__ w2, const float* __restrict__ b2,
    float* __restrict__ out) {

    __shared__ float feats[32 * 8];    // [b][k], k padded 6->8
    __shared__ float w1t[8 * 32];      // [k][n] = w1[n][k]
    __shared__ float hbuf[32 * 32];    // relu(feats @ w1^T + b1)
    __shared__ float w2t[32 * 64];     // [k][j] = w2[j][k]

    const int lane = threadIdx.x;
    const float Nf = (float)(IMG_H * IMG_W);

    for (int idx = lane; idx < 96; idx += 32) {
        int b = idx / 3, ch = idx % 3;
        float S = ws[idx * 2 + 0], S2 = ws[idx * 2 + 1];
        float mean = S / Nf;
        float var  = (S2 - S * S / Nf) / (Nf - 1.0f);   // ddof=1 (unbiased)
        feats[b * 8 + 2 * ch + 0] = mean;
        feats[b * 8 + 2 * ch + 1] = sqrtf(fmaxf(var, 0.0f));
    }
    feats[lane * 8 + 6] = 0.0f;        // K padding
    feats[lane * 8 + 7] = 0.0f;
    for (int i = lane; i < 8 * 32; i += 32) {
        int k = i >> 5, n = i & 31;
        w1t[i] = (k < 6) ? w1[n * 6 + k] : 0.0f;
    }
    for (int i = lane; i < 32 * 64; i += 32) {
        int k = i >> 6, j = i & 63;
        w2t[i] = w2[j * 32 + k];
    }
    __syncthreads();

    const int half = lane >> 4;
    const int r16  = lane & 15;

    // ---- GEMM1 + bias + ReLU ----
    for (int mt = 0; mt < 2; ++mt) {
        for (int nt = 0; nt < 2; ++nt) {
            v8f c = {0.f, 0.f, 0.f, 0.f, 0.f, 0.f, 0.f, 0.f};
#pragma unroll
            for (int k0 = 0; k0 < 8; k0 += 4) {
                int m  = mt * 16 + r16;
                int n  = nt * 16 + r16;
                int kk = k0 + 2 * half;
                v2f a, b;
                a.x = feats[m * 8 + kk];      a.y = feats[m * 8 + kk + 1];
                b.x = w1t[kk * 32 + n];       b.y = w1t[(kk + 1) * 32 + n];
                c = __builtin_amdgcn_wmma_f32_16x16x4_f32(
                        false, a, false, b, (short)0, c, false, false);
            }
            int n = nt * 16 + r16;
#pragma unroll
            for (int v = 0; v < 8; ++v) {
                int m = mt * 16 + half * 8 + v;
                hbuf[m * 32 + n] = fmaxf(c[v] + b1[n], 0.0f);
            }
        }
    }
    __syncthreads();

    // ---- GEMM2 + bias -> d_out [32,64] ----
    for (int mt = 0; mt < 2; ++mt) {
        for (int nt = 0; nt < 4; ++nt) {
            v8f c = {0.f, 0.f, 0.f, 0.f, 0.f, 0.f, 0.f, 0.f};
#pragma unroll
            for (int k0 = 0; k0 < 32; k0 += 4) {
                int m  = mt * 16 + r16;
                int n  = nt * 16 + r16;
                int kk = k0 + 2 * half;
                v2f a, b;
                a.x = hbuf[m * 32 + kk];      a.y = hbuf[m * 32 + kk + 1];
                b.x = w2t[kk * 64 + n];       b.y = w2t[(kk + 1) * 64 + n];
                c = __builtin_amdgcn_wmma_f32_16x16x4_f32(
                        false, a, false, b, (short)0, c, false, false);
            }
            int n = nt * 16 + r16;
#pragma unroll
            for (int v = 0; v < 8; ++v) {
                int m = mt * 16 + half * 8 + v;
                out[m * 64 + n] = c[v] + b2[n];
            }
        }
    }
}

// ---------------------------------------------------------------------------
extern "C" void kernel_launch(void* const* d_in, const int* in_sizes, int n_in,
                              void* d_out, int out_size, void* d_ws, size_t ws_size,
                              hipStream_t stream) {
    const float* x  = (const float*)d_in[0];   // [32,3,512,512]
    const float* w1 = (const float*)d_in[1];   // [32,6]
    const float* b1 = (const float*)d_in[2];   // [32]
    const float* w2 = (const float*)d_in[3];   // [64,32]
    const float* b2 = (const float*)d_in[4];   // [64]
    float* out = (float*)d_out;                // [32,64]
    float* acc = (float*)d_ws;                 // 96 * {sum, sumsq}

    zero_ws_kernel<<<1, 256, 0, stream>>>(acc, 192);
    sobel_pool_stats_kernel<<<96 * TILES_PER_IMG, 256, 0, stream>>>(x, acc);
    mlp_wmma_kernel<<<1, 32, 0, stream>>>(acc, w1, b1, w2, b2, out);
}